// SparseAvgPool2d_5360119185721
// MI455X (gfx1250) — compile-verified
//
#include <hip/hip_runtime.h>
#include <cstdint>

typedef float v4f __attribute__((ext_vector_type(4)));

// Copy one 2KB chunk (4 input rows of 512B) from global to LDS using the
// CDNA5 async global->LDS DMA path (tracked by ASYNCcnt). Each lane moves
// 16B per instruction; the offset immediate is added to BOTH the LDS dest
// address and the global address, so all four issues share operands.
__device__ __forceinline__ void async_copy_2k(unsigned lds_addr, unsigned goff,
                                              unsigned long long gbase) {
  asm volatile(
      "global_load_async_to_lds_b128 %0, %1, %2 offset:0\n\t"
      "global_load_async_to_lds_b128 %0, %1, %2 offset:512\n\t"
      "global_load_async_to_lds_b128 %0, %1, %2 offset:1024\n\t"
      "global_load_async_to_lds_b128 %0, %1, %2 offset:1536"
      :
      : "v"(lds_addr), "v"(goff), "s"(gbase)
      : "memory");
}

// AvgPool2d 2x2 stride 2, fp32, NCHW [32,256,128,128] -> [32,256,64,64].
// Pure bandwidth kernel: 512MB in + 128MB out, ideal ~29us @ 23.3 TB/s.
// One chunk = 4 consecutive input rows (2KB) -> 2 output rows (512B).
__global__ __launch_bounds__(256) void avgpool2x2_kernel(
    const float* __restrict__ x, float* __restrict__ out,
    int num_chunks, int wave_stride) {
  // 8 waves/block * 2 buffers * 512 floats (2KB) = 32KB LDS
  __shared__ float smem[8 * 2 * 512];

  const int lane = threadIdx.x & 31;
  const int wave_in_blk = threadIdx.x >> 5;
  const int gwave = blockIdx.x * 8 + wave_in_blk;

  int chunk = gwave;
  if (chunk >= num_chunks) return;  // wave-uniform

  // Low 32 bits of a flat shared-memory address are the LDS byte offset.
  const unsigned lds_wave = (unsigned)(uintptr_t)(&smem[0]) +
                            (unsigned)(wave_in_blk * 4096) +
                            (unsigned)(lane * 16);
  const unsigned long long gbase = (unsigned long long)(uintptr_t)x;

  // Prologue: fill buffer 0.
  async_copy_2k(lds_wave + 0u, (unsigned)chunk * 2048u + (unsigned)(lane * 16),
                gbase);

  const int pair = lane >> 4;   // 0: output row A, 1: output row B
  const int li = lane & 15;     // position within the output row
  int buf = 0;

  while (true) {
    const int next = chunk + wave_stride;
    const bool more = next < num_chunks;  // wave-uniform

    if (more) {
      // Overlap: start copying the next chunk into the other buffer, then
      // wait until only those 4 newest async loads remain outstanding
      // (async loads complete in order -> current buffer is ready).
      async_copy_2k(lds_wave + (unsigned)((buf ^ 1) * 2048),
                    (unsigned)next * 2048u + (unsigned)(lane * 16), gbase);
      asm volatile("s_wait_asynccnt 4" ::: "memory");
    } else {
      asm volatile("s_wait_asynccnt 0" ::: "memory");
    }

    // Compute 4 outputs per lane from LDS (ds_load_b128 x4).
    const float* s = &smem[wave_in_blk * 1024 + buf * 512 + pair * 256];
    const v4f a0 = *(const v4f*)(s + li * 8);        // row 2p, cols 8li..8li+3
    const v4f a1 = *(const v4f*)(s + li * 8 + 4);    // row 2p, cols 8li+4..8li+7
    const v4f b0 = *(const v4f*)(s + 128 + li * 8);  // row 2p+1
    const v4f b1 = *(const v4f*)(s + 128 + li * 8 + 4);

    v4f r;
    r.x = (a0.x + a0.y + b0.x + b0.y) * 0.25f;
    r.y = (a0.z + a0.w + b0.z + b0.w) * 0.25f;
    r.z = (a1.x + a1.y + b1.x + b1.y) * 0.25f;
    r.w = (a1.z + a1.w + b1.z + b1.w) * 0.25f;

    // Output rows 2*chunk+pair; wave writes one contiguous 512B region.
    const long long orow = (long long)chunk * 2 + pair;
    __builtin_nontemporal_store(r, (v4f*)(out + orow * 64 + li * 4));

    if (!more) break;
    chunk = next;
    buf ^= 1;
  }
}

extern "C" void kernel_launch(void* const* d_in, const int* in_sizes, int n_in,
                              void* d_out, int out_size, void* d_ws,
                              size_t ws_size, hipStream_t stream) {
  const float* x = (const float*)d_in[0];
  float* out = (float*)d_out;

  const long long total = (long long)in_sizes[0];     // 134,217,728 floats
  const int num_chunks = (int)(total / 512);          // 262,144 chunks of 2KB

  int blocks = 2048;                                  // 16,384 waves
  if (blocks * 8 > num_chunks) blocks = (num_chunks + 7) / 8;
  const int wave_stride = blocks * 8;

  avgpool2x2_kernel<<<blocks, 256, 0, stream>>>(x, out, num_chunks,
                                                wave_stride);
}